// GCN_40896678592680
// MI455X (gfx1250) — compile-verified
//
#include <hip/hip_runtime.h>

typedef __attribute__((ext_vector_type(2))) float v2f;
typedef __attribute__((ext_vector_type(8))) float v8f;

// ---------------- small elementwise kernels ----------------

__global__ void zero_f32_kernel(float* __restrict__ p, int n) {
  int i = blockIdx.x * blockDim.x + threadIdx.x;
  if (i < n) p[i] = 0.0f;
}

__global__ void degree_kernel(const int* __restrict__ dst,
                              const float* __restrict__ ew,
                              float* __restrict__ deg, int E) {
  int i = blockIdx.x * blockDim.x + threadIdx.x;
  if (i < E) atomicAdd(&deg[dst[i]], ew[i]);
}

__global__ void dinv_kernel(float* __restrict__ deg, int n) {
  int i = blockIdx.x * blockDim.x + threadIdx.x;
  if (i < n) deg[i] = rsqrtf(deg[i] + 1.0f);  // deg includes self-loop weight 1
}

__global__ void bias_relu_kernel(float* __restrict__ A, const float* __restrict__ b,
                                 int total, int fmask) {
  int i = blockIdx.x * blockDim.x + threadIdx.x;
  if (i < total) {
    float v = A[i] + b[i & fmask];
    A[i] = v > 0.0f ? v : 0.0f;
  }
}

__global__ void bias_out_kernel(const float* __restrict__ A, const float* __restrict__ b,
                                float* __restrict__ out, int total, int fmask) {
  int i = blockIdx.x * blockDim.x + threadIdx.x;
  if (i < total) out[i] = A[i] + b[i & fmask];
}

// Low 32 bits of a flat LDS pointer == workgroup-relative LDS byte offset
// (ISA 10.2: LDS aperture address is truncated to addr[31:0]).
__device__ __forceinline__ unsigned lds_offset_of(const void* p) {
  return (unsigned)(uintptr_t)p;
}

// ---------------- f32 WMMA GEMM: C[n,F] = X[n,K] @ W[K,F] ----------------
// blockDim.x = 32 * (F/16): wave w computes the 16x16 tile at cols [16w,16w+16).
// blockIdx.x selects the 16-row block. Requires n % 16 == 0 (N = 50000 = 3125*16).
// The 16xK A-strip is staged once per block into LDS with the gfx1250 async
// global->LDS copy engine; all waves then read A via ds_load. B (the weight
// matrix, <=64KB) stays in L2 and is read directly.
// Epilogue writes H = result and AGG = result * dinv(row)^2 (self-loop init),
// so AGG needs no prior memset before the edge scatter atomics land on it.
__global__ void gemm_wmma_f32_kernel(const float* __restrict__ X,
                                     const float* __restrict__ W,
                                     const float* __restrict__ dinv,
                                     float* __restrict__ H,
                                     float* __restrict__ AGG,
                                     int K, int F) {
  __shared__ float As[16 * 128];  // K <= 128

  const int wave    = threadIdx.x >> 5;
  const int lane    = threadIdx.x & 31;
  const int half    = lane >> 4;      // 0: lanes 0-15, 1: lanes 16-31
  const int lane16  = lane & 15;
  const int rowBase = blockIdx.x << 4;
  const int colBase = wave << 4;
  const int col     = colBase + lane16;  // B/C/D column held by this lane

  // ---- async stage A[rowBase:rowBase+16, 0:K] -> LDS (16B per lane-op) ----
  // total chunks = 16*K/4 float4s; blockDim divides it exactly (512 / {128,256}),
  // so every lane stays active (EXEC all-ones throughout, as WMMA requires).
  const int nChunks = (16 * K) >> 2;
  for (int c = threadIdx.x; c < nChunks; c += blockDim.x) {
    const unsigned ldsOff = lds_offset_of(&As[c << 2]);
    const float* g = X + (size_t)rowBase * K + ((size_t)c << 2);
    asm volatile("global_load_async_to_lds_b128 %0, %1, off"
                 :: "v"(ldsOff), "v"(g)
                 : "memory");
  }
  asm volatile("s_wait_asynccnt 0x0" ::: "memory");
  __syncthreads();

  const float* __restrict__ arow = As + lane16 * K;

  v8f acc = {};
  for (int k = 0; k < K; k += 4) {
    // 32-bit A 16x4 layout (ISA 7.12.2): lanes 0-15 hold K=k,k+1 in VGPR0/1,
    // lanes 16-31 hold K=k+2,k+3.  (ka even -> 8B-aligned ds_load_b64)
    const int ka = k + (half << 1);
    const v2f a = *(const v2f*)(arow + ka);
    // B 4x16 (KxN), mirrored: N on lanes, K split across lane-halves/VGPRs.
    v2f bb;
    bb.x = W[(size_t)ka * F + col];
    bb.y = W[(size_t)(ka + 1) * F + col];
    acc = __builtin_amdgcn_wmma_f32_16x16x4_f32(false, a, false, bb,
                                                (short)0, acc, false, false);
  }

  // C/D layout: VGPR v, lanes 0-15 -> M=v, lanes 16-31 -> M=v+8; N = lane16.
#pragma unroll
  for (int v = 0; v < 8; ++v) {
    const int m = rowBase + (half << 3) + v;
    const float val = acc[v];
    const float di = dinv[m];
    const size_t idx = (size_t)m * F + col;
    H[idx]   = val;
    AGG[idx] = val * (di * di);
  }
}

// ---------------- edge scatter: AGG[dst] += H[src] * norm ----------------
// One wave per edge; lane owns 4 contiguous features (F = 128).
__global__ void scatter128_kernel(const int* __restrict__ src, const int* __restrict__ dst,
                                  const float* __restrict__ ew, const float* __restrict__ dinv,
                                  const float* __restrict__ H, float* __restrict__ AGG, int E) {
  const int lane = threadIdx.x & 31;
  int w = (blockIdx.x * blockDim.x + threadIdx.x) >> 5;
  const int nw = (gridDim.x * blockDim.x) >> 5;
  for (int e = w; e < E; e += nw) {
    const int s = src[e];
    const int d = dst[e];
    const float nrm = dinv[s] * ew[e] * dinv[d];
    const float4 v = *(const float4*)(H + (size_t)s * 128 + lane * 4);
    float* o = AGG + (size_t)d * 128 + lane * 4;
    atomicAdd(o + 0, v.x * nrm);
    atomicAdd(o + 1, v.y * nrm);
    atomicAdd(o + 2, v.z * nrm);
    atomicAdd(o + 3, v.w * nrm);
  }
}

// One wave per edge; lane owns 2 contiguous features (F = 64).
__global__ void scatter64_kernel(const int* __restrict__ src, const int* __restrict__ dst,
                                 const float* __restrict__ ew, const float* __restrict__ dinv,
                                 const float* __restrict__ H, float* __restrict__ AGG, int E) {
  const int lane = threadIdx.x & 31;
  int w = (blockIdx.x * blockDim.x + threadIdx.x) >> 5;
  const int nw = (gridDim.x * blockDim.x) >> 5;
  for (int e = w; e < E; e += nw) {
    const int s = src[e];
    const int d = dst[e];
    const float nrm = dinv[s] * ew[e] * dinv[d];
    const float2 v = *(const float2*)(H + (size_t)s * 64 + lane * 2);
    float* o = AGG + (size_t)d * 64 + lane * 2;
    atomicAdd(o + 0, v.x * nrm);
    atomicAdd(o + 1, v.y * nrm);
  }
}

// ---------------- driver ----------------

extern "C" void kernel_launch(void* const* d_in, const int* in_sizes, int n_in,
                              void* d_out, int out_size, void* d_ws, size_t ws_size,
                              hipStream_t stream) {
  (void)n_in; (void)out_size; (void)ws_size;

  const float* x  = (const float*)d_in[0];   // [N,128]
  const int*   ei = (const int*)d_in[1];     // [2,E]
  const float* ew = (const float*)d_in[2];   // [E,1]
  const float* W1 = (const float*)d_in[3];   // [128,128]
  const float* b1 = (const float*)d_in[4];   // [128]
  const float* W2 = (const float*)d_in[5];   // [128,64]
  const float* b2 = (const float*)d_in[6];   // [64]
  float* out = (float*)d_out;                // [N,64]

  const int F_IN = 128, F_H = 128, F_OUT = 64;
  const int N = in_sizes[0] / F_IN;
  const int E = in_sizes[1] / 2;
  const int* srcIdx = ei;       // edge_index[0]
  const int* dstIdx = ei + E;   // edge_index[1]

  // Workspace layout (floats): dinv | h1 | agg1 | h2 | agg2
  float* ws = (float*)d_ws;
  size_t off = 0;
  float* dinv = ws + off;  off += (size_t)((N + 255) & ~255);
  float* h1   = ws + off;  off += (size_t)N * F_H;
  float* agg1 = ws + off;  off += (size_t)N * F_H;
  float* h2   = ws + off;  off += (size_t)N * F_OUT;
  float* agg2 = ws + off;

  // 1) degrees -> dinv (in place)
  zero_f32_kernel<<<(N + 255) / 256, 256, 0, stream>>>(dinv, N);
  degree_kernel<<<(E + 255) / 256, 256, 0, stream>>>(dstIdx, ew, dinv, E);
  dinv_kernel<<<(N + 255) / 256, 256, 0, stream>>>(dinv, N);

  // 2) layer 1: GEMM (+self-loop init), scatter, bias+relu (in place on agg1)
  gemm_wmma_f32_kernel<<<N / 16, 32 * (F_H / 16), 0, stream>>>(x, W1, dinv, h1, agg1, F_IN, F_H);
  scatter128_kernel<<<4096, 256, 0, stream>>>(srcIdx, dstIdx, ew, dinv, h1, agg1, E);
  bias_relu_kernel<<<(N * F_H + 255) / 256, 256, 0, stream>>>(agg1, b1, N * F_H, F_H - 1);

  // 3) layer 2: GEMM on agg1 (=h), scatter, bias -> out
  gemm_wmma_f32_kernel<<<N / 16, 32 * (F_OUT / 16), 0, stream>>>(agg1, W2, dinv, h2, agg2, F_H, F_OUT);
  scatter64_kernel<<<4096, 256, 0, stream>>>(srcIdx, dstIdx, ew, dinv, h2, agg2, E);
  bias_out_kernel<<<(N * F_OUT + 255) / 256, 256, 0, stream>>>(agg2, b2, out, N * F_OUT, F_OUT - 1);
}